// HybridLoss_53334903882250
// MI455X (gfx1250) — compile-verified
//
#include <hip/hip_runtime.h>
#include <math.h>

#define TPB 64

typedef __attribute__((ext_vector_type(2))) float v2f;
typedef __attribute__((ext_vector_type(8))) float v8f;

// Masks 1..255 grouped by popcount so the pred-corner row r is compile-time
// after unrolling -> cost[r][j] is statically indexed and stays in VGPRs.
struct MaskTable {
  unsigned char m[255];
  short off[9];
};

constexpr MaskTable makeTable() {
  MaskTable t{};
  int p = 0;
  for (int r = 1; r <= 8; ++r) {
    t.off[r - 1] = (short)p;
    for (int mask = 1; mask < 256; ++mask) {
      int c = 0;
      for (int b = 0; b < 8; ++b) c += (mask >> b) & 1;
      if (c == r) t.m[p++] = (unsigned char)mask;
    }
  }
  t.off[8] = (short)p;
  return t;
}

__constant__ MaskTable kMasks = makeTable();

__device__ __forceinline__ void quatToMat(float x, float y, float z, float w,
                                          float R[3][3]) {
  R[0][0] = 1.f - 2.f * (y * y + z * z);
  R[0][1] = 2.f * (x * y - z * w);
  R[0][2] = 2.f * (x * z + y * w);
  R[1][0] = 2.f * (x * y + z * w);
  R[1][1] = 1.f - 2.f * (x * x + z * z);
  R[1][2] = 2.f * (y * z - x * w);
  R[2][0] = 2.f * (x * z - y * w);
  R[2][1] = 2.f * (y * z + x * w);
  R[2][2] = 1.f - 2.f * (x * x + y * y);
}

// dot(col ja of A, col jb of B); dynamic column resolved via selects so the
// matrices never leave registers.
__device__ __forceinline__ float colDot(const float A[3][3], int ja,
                                        const float B[3][3], int jb) {
  float a0 = (ja == 0) ? A[0][0] : ((ja == 1) ? A[0][1] : A[0][2]);
  float a1 = (ja == 0) ? A[1][0] : ((ja == 1) ? A[1][1] : A[1][2]);
  float a2 = (ja == 0) ? A[2][0] : ((ja == 1) ? A[2][1] : A[2][2]);
  float b0 = (jb == 0) ? B[0][0] : ((jb == 1) ? B[0][1] : B[0][2]);
  float b1 = (jb == 0) ? B[1][0] : ((jb == 1) ? B[1][1] : B[1][2]);
  float b2 = (jb == 0) ? B[2][0] : ((jb == 1) ? B[2][1] : B[2][2]);
  return a0 * b0 + a1 * b1 + a2 * b2;
}

__global__ void __launch_bounds__(TPB) hybrid_loss_main(
    const float* __restrict__ pp, const float* __restrict__ gp,
    float* __restrict__ partial, int nBox) {
  // dp[mask][tid]: 64KB, lanes on consecutive banks, mask offset is scalar.
  __shared__ float dp[256 * TPB];

  const int tid = threadIdx.x;
  const int box = blockIdx.x * TPB + tid;
  const bool active = box < nBox;

  float P[10], G[10];
#pragma unroll
  for (int k = 0; k < 10; ++k) { P[k] = 1.0f; G[k] = 1.0f; }
  if (active) {
    const float2* p2 = reinterpret_cast<const float2*>(pp + (size_t)box * 10);
    const float2* g2 = reinterpret_cast<const float2*>(gp + (size_t)box * 10);
#pragma unroll
    for (int k = 0; k < 5; ++k) {
      float2 a = p2[k];
      float2 b = g2[k];
      P[2 * k] = a.x; P[2 * k + 1] = a.y;
      G[2 * k] = b.x; G[2 * k + 1] = b.y;
    }
  }

  const float pcen[3] = {P[0], P[1], P[2]};
  const float gcen[3] = {G[0], G[1], G[2]};
  const float pd[3] = {P[3], P[4], P[5]};
  const float gd[3] = {G[3], G[4], G[5]};
  const float pq[4] = {P[6], P[7], P[8], P[9]};
  const float gq[4] = {G[6], G[7], G[8], G[9]};

  const float gdiag = sqrtf(gd[0] * gd[0] + gd[1] * gd[1] + gd[2] * gd[2]);

  // ---- diag + center ----
  float diagT = 0.f, centerT = 0.f;
#pragma unroll
  for (int k = 0; k < 3; ++k) {
    diagT += fabsf(pd[k] - gd[k]) / (gd[k] + 1e-6f);
    centerT += fabsf(pcen[k] - gcen[k]);
  }
  diagT *= (1.f / 3.f);
  centerT *= (1.f / 3.f) / gdiag;

  const float pnrm =
      sqrtf(pq[0] * pq[0] + pq[1] * pq[1] + pq[2] * pq[2] + pq[3] * pq[3]);
  const float gnrm =
      sqrtf(gq[0] * gq[0] + gq[1] * gq[1] + gq[2] * gq[2] + gq[3] * gq[3]);

  // ---- quat loss (eps = 1e-6) ----
  float quatT = 0.f;
  {
    const float ip6 = 1.f / (pnrm + 1e-6f);
    const float ig6 = 1.f / (gnrm + 1e-6f);
    float Rp6[3][3], Rg6[3][3];
    quatToMat(pq[0] * ip6, pq[1] * ip6, pq[2] * ip6, pq[3] * ip6, Rp6);
    quatToMat(gq[0] * ig6, gq[1] * ig6, gq[2] * ig6, gq[3] * ig6, Rg6);
#pragma unroll
    for (int i = 0; i < 3; ++i) {
      float mx = 0.f;
#pragma unroll
      for (int j = 0; j < 3; ++j) {
        float d = Rp6[0][i] * Rg6[0][j] + Rp6[1][i] * Rg6[1][j] +
                  Rp6[2][i] * Rg6[2][j];
        mx = fmaxf(mx, fabsf(d));
      }
      mx = fminf(fmaxf(mx, 1e-6f), 1.0f);
      quatT += acosf(mx);
    }
    quatT *= (1.f / 3.f);
  }

  // ---- rotation matrices with eps = 1e-8 (corners + canonical) ----
  float Rp[3][3], Rg[3][3];
  {
    const float ip8 = 1.f / (pnrm + 1e-8f);
    const float ig8 = 1.f / (gnrm + 1e-8f);
    quatToMat(pq[0] * ip8, pq[1] * ip8, pq[2] * ip8, pq[3] * ip8, Rp);
    quatToMat(gq[0] * ig8, gq[1] * ig8, gq[2] * ig8, gq[3] * ig8, Rg);
  }

  // ---- canonical loss ----
  float canonT;
  {
    float ps0 = pd[0], ps1 = pd[1], ps2 = pd[2];
    int pi0 = 0, pi1 = 1, pi2 = 2;
    if (ps0 < ps1) { float t = ps0; ps0 = ps1; ps1 = t; int u = pi0; pi0 = pi1; pi1 = u; }
    if (ps1 < ps2) { float t = ps1; ps1 = ps2; ps2 = t; int u = pi1; pi1 = pi2; pi2 = u; }
    if (ps0 < ps1) { float t = ps0; ps0 = ps1; ps1 = t; int u = pi0; pi0 = pi1; pi1 = u; }
    float gs0 = gd[0], gs1 = gd[1], gs2 = gd[2];
    int gi0 = 0, gi1 = 1, gi2 = 2;
    if (gs0 < gs1) { float t = gs0; gs0 = gs1; gs1 = t; int u = gi0; gi0 = gi1; gi1 = u; }
    if (gs1 < gs2) { float t = gs1; gs1 = gs2; gs2 = t; int u = gi1; gi1 = gi2; gi2 = u; }
    if (gs0 < gs1) { float t = gs0; gs0 = gs1; gs1 = t; int u = gi0; gi0 = gi1; gi1 = u; }

    const float dimsT = (fabsf(ps0 - gs0) / (gs0 + 1e-8f) +
                         fabsf(ps1 - gs1) / (gs1 + 1e-8f) +
                         fabsf(ps2 - gs2) / (gs2 + 1e-8f)) * (1.f / 3.f);

    const int pinv0 = (pi0 == 0) ? 0 : ((pi1 == 0) ? 1 : 2);
    const int pinv1 = (pi0 == 1) ? 0 : ((pi1 == 1) ? 1 : 2);
    const int pinv2 = (pi0 == 2) ? 0 : ((pi1 == 2) ? 1 : 2);
    const int ginv0 = (gi0 == 0) ? 0 : ((gi1 == 0) ? 1 : 2);
    const int ginv1 = (gi0 == 1) ? 0 : ((gi1 == 1) ? 1 : 2);
    const int ginv2 = (gi0 == 2) ? 0 : ((gi1 == 2) ? 1 : 2);

    float angT = 0.f;
    angT += acosf(fminf(fmaxf(fabsf(colDot(Rp, pinv0, Rg, ginv0)), 0.f), 1.f));
    angT += acosf(fminf(fmaxf(fabsf(colDot(Rp, pinv1, Rg, ginv1)), 0.f), 1.f));
    angT += acosf(fminf(fmaxf(fabsf(colDot(Rp, pinv2, Rg, ginv2)), 0.f), 1.f));
    angT *= (1.f / 3.f);
    canonT = 0.5f * dimsT + 0.5f * angT;
  }

  // ---- corners ----
  float Mp[3][3], Mg[3][3];
#pragma unroll
  for (int i = 0; i < 3; ++i)
#pragma unroll
    for (int j = 0; j < 3; ++j) {
      Mp[i][j] = Rp[i][j] * (0.5f * pd[j]);
      Mg[i][j] = Rg[i][j] * (0.5f * gd[j]);
    }

  float pcs[8][3], gcs[8][3];
#pragma unroll
  for (int k = 0; k < 8; ++k) {
    const float sx = (k & 4) ? -1.f : 1.f;
    const float sy = (k & 2) ? -1.f : 1.f;
    const float sz = (k & 1) ? -1.f : 1.f;
#pragma unroll
    for (int i = 0; i < 3; ++i) {
      pcs[k][i] = pcen[i] + sx * Mp[i][0] + sy * Mp[i][1] + sz * Mp[i][2];
      gcs[k][i] = gcen[i] + sx * Mg[i][0] + sy * Mg[i][1] + sz * Mg[i][2];
    }
  }

  // ---- per-axis AABBs (shared by all three IoU planes) ----
  float pmin[3], pmax[3], gmin[3], gmax[3];
#pragma unroll
  for (int i = 0; i < 3; ++i) {
    float pmn = pcs[0][i], pmx = pcs[0][i], gmn = gcs[0][i], gmx = gcs[0][i];
#pragma unroll
    for (int k = 1; k < 8; ++k) {
      pmn = fminf(pmn, pcs[k][i]); pmx = fmaxf(pmx, pcs[k][i]);
      gmn = fminf(gmn, gcs[k][i]); gmx = fmaxf(gmx, gcs[k][i]);
    }
    pmin[i] = pmn; pmax[i] = pmx; gmin[i] = gmn; gmax[i] = gmx;
  }

  // ---- 8x8 L1 cost ----
  float cost[8][8];
#pragma unroll
  for (int r = 0; r < 8; ++r)
#pragma unroll
    for (int j = 0; j < 8; ++j)
      cost[r][j] = fabsf(pcs[r][0] - gcs[j][0]) + fabsf(pcs[r][1] - gcs[j][1]) +
                   fabsf(pcs[r][2] - gcs[j][2]);

  // ---- bitmask-DP assignment: per-thread LDS column, no syncs needed ----
  dp[tid] = 0.f;  // mask 0
#pragma unroll
  for (int r = 0; r < 8; ++r) {
    const int lo = kMasks.off[r];
    const int hi = kMasks.off[r + 1];
    for (int idx = lo; idx < hi; ++idx) {
      const int mask = (int)kMasks.m[idx];
      float best = 3.0e38f;
#pragma unroll
      for (int j = 0; j < 8; ++j) {
        if (mask & (1 << j)) {
          best = fminf(best, dp[(mask ^ (1 << j)) * TPB + tid] + cost[r][j]);
        }
      }
      dp[mask * TPB + tid] = best;
    }
  }
  const float cornerT = dp[255 * TPB + tid] * 0.125f / (gdiag + 1e-6f);

  // ---- projected IoU ----
  float iouSum = 0.f;
  const int A0[3] = {0, 0, 1}, A1[3] = {2, 1, 2}, HH[3] = {1, 2, 0};
#pragma unroll
  for (int p = 0; p < 3; ++p) {
    const int a = A0[p], b = A1[p], h = HH[p];
    float ia = fmaxf(fminf(pmax[a], gmax[a]) - fmaxf(pmin[a], gmin[a]) + 1e-3f, 0.f);
    float ib = fmaxf(fminf(pmax[b], gmax[b]) - fmaxf(pmin[b], gmin[b]) + 1e-3f, 0.f);
    float inter = ia * ib;
    float a1 = (pmax[a] - pmin[a] + 1e-3f) * (pmax[b] - pmin[b] + 1e-3f);
    float a2 = (gmax[a] - gmin[a] + 1e-3f) * (gmax[b] - gmin[b] + 1e-3f);
    float iou2 = inter / (a1 + a2 - inter + 1e-6f);
    float hi2 = fmaxf(fminf(pmax[h], gmax[h]) - fmaxf(pmin[h], gmin[h]), 0.f);
    float hu = (pmax[h] - pmin[h]) + (gmax[h] - gmin[h]) - hi2;
    iouSum += iou2 * (hi2 / (hu + 1e-6f));
  }
  const float iouB = fminf(fmaxf(iouSum * (1.f / 3.f), 0.f), 1.f);

  float contrib = 0.15f * diagT + 0.2f * centerT + 0.15f * quatT +
                  0.1f * canonT + 0.2f * cornerT + 0.2f * (1.f - iouB);
  if (!active) contrib = 0.f;

  // ---- wave32 + block reduction (reuse dp LDS after barrier) ----
  float s = contrib;
#pragma unroll
  for (int o = 16; o > 0; o >>= 1) s += __shfl_down(s, o, 32);
  __syncthreads();
  if ((tid & 31) == 0) dp[tid >> 5] = s;
  __syncthreads();
  if (tid == 0) partial[blockIdx.x] = dp[0] + dp[1];
}

// WMMA-based final reduction: D = A * ones(4x16) + C replicates the 16 row
// sums across all 16 columns; sum of all 256 C/D slots == 16 * sum(values).
__global__ void __launch_bounds__(32) hybrid_loss_reduce(
    const float* __restrict__ ws, int n, float* __restrict__ out, float scale) {
  const int lane = threadIdx.x;  // exactly one wave32 -> EXEC all ones
  float s = 0.f;
#if __has_builtin(__builtin_amdgcn_wmma_f32_16x16x4_f32)
  const int nFull = n & ~63;
  v8f acc = {};
  v2f ones;
  ones.x = 1.f;
  ones.y = 1.f;
  for (int base = 0; base < nFull; base += 64) {
    v2f a;
    a.x = ws[base + 2 * lane];
    a.y = ws[base + 2 * lane + 1];
    // 8 args: (neg_a, A, neg_b, B, c_mod, C, reuse_a, reuse_b)
    acc = __builtin_amdgcn_wmma_f32_16x16x4_f32(false, a, false, ones,
                                                (short)0, acc, false, false);
  }
#pragma unroll
  for (int v = 0; v < 8; ++v) s += acc[v];
  s *= (1.f / 16.f);
  for (int idx = nFull + lane; idx < n; idx += 32) s += ws[idx];
#else
  for (int idx = lane; idx < n; idx += 32) s += ws[idx];
#endif
#pragma unroll
  for (int o = 16; o > 0; o >>= 1) s += __shfl_down(s, o, 32);
  if (lane == 0) out[0] = s * scale;
}

extern "C" void kernel_launch(void* const* d_in, const int* in_sizes, int n_in,
                              void* d_out, int out_size, void* d_ws,
                              size_t ws_size, hipStream_t stream) {
  const float* pp = (const float*)d_in[0];
  const float* gp = (const float*)d_in[1];
  const int nBox = in_sizes[0] / 10;      // B = 131072
  const int blocks = (nBox + TPB - 1) / TPB;  // 2048 partials (8 KB of d_ws)
  float* ws = (float*)d_ws;
  hybrid_loss_main<<<blocks, TPB, 0, stream>>>(pp, gp, ws, nBox);
  hybrid_loss_reduce<<<1, 32, 0, stream>>>(ws, blocks, (float*)d_out,
                                           1.0f / (float)nBox);
}